// TFNConv_26079041421317
// MI455X (gfx1250) — compile-verified
//
#include <hip/hip_runtime.h>
#include <math.h>

typedef float v2f __attribute__((ext_vector_type(2)));
typedef float v8f __attribute__((ext_vector_type(8)));

__device__ __forceinline__ v8f wmma4(v2f a, v2f b, v8f c) {
    // D = A(16x4) * B(4x16) + C, fp32, wave32
    return __builtin_amdgcn_wmma_f32_16x16x4_f32(
        false, a, false, b, (short)0, c, false, false);
}

__device__ __forceinline__ v2f ld2(const float* p) {   // aligned 8B pair load
    return *(const v2f*)p;
}

__device__ __forceinline__ void atomAddF(float* p, float v) {
    unsafeAtomicAdd(p, v);   // global_atomic_add_f32 on gfx1250
}

// pair-interleaved index: element (k, col) of a KxCOLS matrix stored so that
// rows k0 (even) and k0+1 for one col are adjacent -> single b64 load for WMMA B
#define PIDX(k, col, COLS) ((((k) >> 1) * (COLS) + (col)) * 2 + ((k) & 1))

// ---------------------------------------------------------------------------
// Kernel 0: pre-transpose Wsc_s / Wsc_v (512x32) into pair layout, x fc folded
// ---------------------------------------------------------------------------
__global__ __launch_bounds__(256) void k_prep_wsc(
    const float* __restrict__ Wsc_s, const float* __restrict__ Wsc_v,
    float* __restrict__ wsp_s, float* __restrict__ wsp_v)
{
    const float fc = 0.044194173824159216f;    // 1/sqrt(512)
    const int i = blockIdx.x * 256 + threadIdx.x;   // i in [0, 16384)
    if (i < 512 * 32) {
        const int k = i >> 5, col = i & 31;
        wsp_s[PIDX(k, col, 32)] = Wsc_s[i] * fc;
        wsp_v[PIDX(k, col, 32)] = Wsc_v[i] * fc;
    }
}

// ---------------------------------------------------------------------------
// Kernel 1: s = s_in @ W1s / sqrt(M);  v[:,:,c] = v_in[:,:,c] @ W1v / sqrt(M)
// sv[n][0:32]=s, sv[n][32 + k*3 + c] = v[n][k][c]  (gather-friendly)
// ---------------------------------------------------------------------------
__global__ __launch_bounds__(128) void k_node_pre(
    const float* __restrict__ nf, const float* __restrict__ W1s,
    const float* __restrict__ W1v, float* __restrict__ sv, int N)
{
    const float inv_m = 0.17677669529663687f;  // 1/sqrt(32), folded into weights
    __shared__ float lw[2048];                 // [0:1024) W1s, [1024:2048) W1v, pair layout
    for (int i = threadIdx.x; i < 1024; i += 128) {
        const int k = i >> 5, col = i & 31;
        lw[PIDX(k, col, 32)]        = W1s[i] * inv_m;
        lw[1024 + PIDX(k, col, 32)] = W1v[i] * inv_m;
    }
    __syncthreads();

    const int lane = threadIdx.x & 31;
    const int wv   = threadIdx.x >> 5;
    const int tile = blockIdx.x * 4 + wv;
    const int ntiles = (N + 15) / 16;
    if (tile >= ntiles) return;
    const int tb   = tile * 16;
    const int rl   = lane & 15;
    const int half = lane >> 4;
    const int kb   = half * 2;
    const int arow = min(tb + rl, N - 1);
    const bool full = (tb + 16 <= N);

    {   // scalar channel
        v8f c0 = {}; v8f c1 = {};
        #pragma unroll
        for (int j = 0; j < 8; ++j) {
            const int k0 = 4 * j + kb;         // even
            const int kp = (k0 >> 1) * 64;     // pair-row base (*32 cols *2)
            v2f a  = ld2(&nf[arow * 128 + k0]);
            v2f b0 = ld2(&lw[kp + rl * 2]);
            v2f b1 = ld2(&lw[kp + (16 + rl) * 2]);
            c0 = wmma4(a, b0, c0);
            c1 = wmma4(a, b1, c1);
        }
        if (full) {
            #pragma unroll
            for (int r = 0; r < 8; ++r) {
                const int node = tb + r + 8 * half;
                sv[node * 128 + rl]      = c0[r];
                sv[node * 128 + 16 + rl] = c1[r];
            }
        } else {
            #pragma unroll
            for (int r = 0; r < 8; ++r) {
                const int node = tb + r + 8 * half;
                if (node < N) {
                    sv[node * 128 + rl]      = c0[r];
                    sv[node * 128 + 16 + rl] = c1[r];
                }
            }
        }
    }
    #pragma unroll
    for (int cc = 0; cc < 3; ++cc) {           // vector channels
        v8f c0 = {}; v8f c1 = {};
        #pragma unroll
        for (int j = 0; j < 8; ++j) {
            const int k0 = 4 * j + kb;
            const int kp = (k0 >> 1) * 64;
            v2f a;
            a.x = nf[arow * 128 + 32 + k0 * 3 + cc];
            a.y = nf[arow * 128 + 32 + (k0 + 1) * 3 + cc];
            v2f b0 = ld2(&lw[1024 + kp + rl * 2]);
            v2f b1 = ld2(&lw[1024 + kp + (16 + rl) * 2]);
            c0 = wmma4(a, b0, c0);
            c1 = wmma4(a, b1, c1);
        }
        if (full) {
            #pragma unroll
            for (int r = 0; r < 8; ++r) {
                const int node = tb + r + 8 * half;
                sv[node * 128 + 32 + rl * 3 + cc]        = c0[r];
                sv[node * 128 + 32 + (16 + rl) * 3 + cc] = c1[r];
            }
        } else {
            #pragma unroll
            for (int r = 0; r < 8; ++r) {
                const int node = tb + r + 8 * half;
                if (node < N) {
                    sv[node * 128 + 32 + rl * 3 + cc]        = c0[r];
                    sv[node * 128 + 32 + (16 + rl) * 3 + cc] = c1[r];
                }
            }
        }
    }
}

// ---------------------------------------------------------------------------
// Edge scatter: per-lane message computation + fp32 atomic scatter
// ---------------------------------------------------------------------------
template <bool CHK>
__device__ __forceinline__ void scatter16(
    int eb, int half, int rl, int E,
    const int* __restrict__ eidx, const float* __restrict__ eattr,
    const float* __restrict__ sv, float* __restrict__ acc, const v8f* cw)
{
    const float IS3 = 0.5773502691896258f;   // 1/sqrt(3)
    #pragma unroll
    for (int r = 0; r < 8; ++r) {
        const int e = eb + 8 * half + r;
        if (!CHK || e < E) {
            const int src = eidx[e];
            const int dst = eidx[E + e];
            const float4 ea = *(const float4*)&eattr[e * 4];
            const float e0 = ea.x, e1x = ea.y, e1y = ea.z, e1z = ea.w;
            const float* svp = sv + (size_t)src * 128;
            float* ap = acc + (size_t)dst * 256;
            #pragma unroll
            for (int hm = 0; hm < 2; ++hm) {
                const int m = 16 * hm + rl;
                const float w0 = cw[0 + hm][r];
                const float w1 = cw[2 + hm][r];
                const float w2 = cw[4 + hm][r];
                const float w3 = cw[6 + hm][r];
                const float ssv = svp[m];
                const float v0  = svp[32 + m * 3 + 0];
                const float v1  = svp[32 + m * 3 + 1];
                const float v2  = svp[32 + m * 3 + 2];
                atomAddF(ap + m, w0 * ssv * e0);                        // m0
                const float dot = v0 * e1x + v1 * e1y + v2 * e1z;
                atomAddF(ap + 32 + m, w3 * dot * IS3);                  // m3
                const float c1 = w1 * ssv;                              // m1
                atomAddF(ap + 64 + m * 3 + 0, c1 * e1x);
                atomAddF(ap + 64 + m * 3 + 1, c1 * e1y);
                atomAddF(ap + 64 + m * 3 + 2, c1 * e1z);
                const float c2 = w2 * e0;                               // m2
                atomAddF(ap + 64 + (32 + m) * 3 + 0, c2 * v0);
                atomAddF(ap + 64 + (32 + m) * 3 + 1, c2 * v1);
                atomAddF(ap + 64 + (32 + m) * 3 + 2, c2 * v2);
            }
        }
    }
}

// ---------------------------------------------------------------------------
// Kernel 2 (persistent): per 16-edge tile:
//   h = ssp(emb @ Wr0 / sqrt(8));  w = h @ Wr1 / 8  (both via WMMA)
//   then message computation + fp32 atomic scatter into acc[N][256]
// ---------------------------------------------------------------------------
__global__ __launch_bounds__(128) void k_edge(
    const float* __restrict__ emb, const float* __restrict__ eattr,
    const int* __restrict__ eidx,
    const float* __restrict__ Wr0, const float* __restrict__ Wr1,
    const float* __restrict__ sv, float* __restrict__ acc, int E)
{
    const float invR = 0.3535533905932738f;   // 1/sqrt(8), folded into Wr0
    const float invH = 0.125f;                // 1/sqrt(64), folded into Wr1
    __shared__ float lw0[512];       // Wr0 8x64, pair layout
    __shared__ float lw1[8192];      // Wr1 64x128, pair layout
    __shared__ float lh[4][1024];    // per-wave h staging, 16x64 row-major
    for (int i = threadIdx.x; i < 512; i += 128) {
        const int k = i >> 6, col = i & 63;
        lw0[PIDX(k, col, 64)] = Wr0[i] * invR;
    }
    for (int i = threadIdx.x; i < 8192; i += 128) {
        const int k = i >> 7, col = i & 127;
        lw1[PIDX(k, col, 128)] = Wr1[i] * invH;
    }
    __syncthreads();

    const int lane = threadIdx.x & 31;
    const int wv   = threadIdx.x >> 5;
    const int rl   = lane & 15;
    const int half = lane >> 4;
    const int kb   = half * 2;
    const int ntiles = (E + 15) / 16;
    const float LOG2C = 0.6931471805599453f;

    for (int tile = blockIdx.x * 4 + wv; tile < ntiles; tile += gridDim.x * 4) {
        const int eb    = tile * 16;
        const int aedge = min(eb + rl, E - 1);

        const int nt = tile + gridDim.x * 4;
        if (nt < ntiles) __builtin_prefetch(&emb[(nt * 16 + rl) * 8], 0, 1);

        // ---- stage 1: h = ssp(emb @ Wr0') ----
        v8f ch[4]; ch[0] = {}; ch[1] = {}; ch[2] = {}; ch[3] = {};
        #pragma unroll
        for (int j = 0; j < 2; ++j) {
            const int k0 = 4 * j + kb;
            const int kp = (k0 >> 1) * 128;    // *64 cols *2
            v2f a = ld2(&emb[aedge * 8 + k0]);
            #pragma unroll
            for (int t = 0; t < 4; ++t) {
                v2f b = ld2(&lw0[kp + (t * 16 + rl) * 2]);
                ch[t] = wmma4(a, b, ch[t]);
            }
        }
        // ssp + C-layout -> row-major LDS (A-layout source for stage 2)
        #pragma unroll
        for (int t = 0; t < 4; ++t) {
            #pragma unroll
            for (int r = 0; r < 8; ++r) {
                const float x  = ch[t][r];
                const float tt = __expf(-fabsf(x));
                const float sp = fmaxf(x, 0.0f) + __logf(1.0f + tt);
                lh[wv][(r + 8 * half) * 64 + t * 16 + rl] = sp - LOG2C;
            }
        }
        __builtin_amdgcn_wave_barrier();

        // ---- stage 2: w = h @ Wr1'  (8 col tiles x 16 k-steps) ----
        v8f cw[8];
        #pragma unroll
        for (int t = 0; t < 8; ++t) cw[t] = {};
        #pragma unroll 2
        for (int j = 0; j < 16; ++j) {
            const int k0 = 4 * j + kb;
            const int kp = (k0 >> 1) * 256;    // *128 cols *2
            v2f a = ld2(&lh[wv][rl * 64 + k0]);
            #pragma unroll
            for (int t = 0; t < 8; ++t) {
                v2f b = ld2(&lw1[kp + (t * 16 + rl) * 2]);
                cw[t] = wmma4(a, b, cw[t]);
            }
        }

        // ---- stage 3: messages + atomic scatter ----
        if (eb + 16 <= E) {
            scatter16<false>(eb, half, rl, E, eidx, eattr, sv, acc, cw);
        } else {
            scatter16<true>(eb, half, rl, E, eidx, eattr, sv, acc, cw);
        }
    }
}

// ---------------------------------------------------------------------------
// Kernel 3: out_s = ns@W2s*alpha + (s_in (x) attrs)@Wsc_s*fc     (K=64 / K=512)
//           out_v[c] = nv[c]@W2v*alpha + (v_in[c] (x) attrs)@Wsc_v*fc
// (alpha folded into staged W2s/W2v; fc folded into wsp_s/wsp_v by k_prep_wsc)
// ---------------------------------------------------------------------------
__global__ __launch_bounds__(128) void k_node_post(
    const float* __restrict__ nf, const float* __restrict__ attrs,
    const float* __restrict__ acc,
    const float* __restrict__ W2s, const float* __restrict__ W2v,
    const float* __restrict__ wsp_s, const float* __restrict__ wsp_v,
    float* __restrict__ out, int N)
{
    const float alpha = 0.03125f;          // 1/sqrt(64) * 1/sqrt(16)
    __shared__ float lw2s[2048];           // 64x32, pair layout
    __shared__ float lw2v[2048];
    __shared__ float lnode[4][16 * 144];   // per-wave: [row][0:128 nf | 128:144 attrs]
    for (int i = threadIdx.x; i < 2048; i += 128) {
        const int k = i >> 5, col = i & 31;
        lw2s[PIDX(k, col, 32)] = W2s[i] * alpha;
        lw2v[PIDX(k, col, 32)] = W2v[i] * alpha;
    }
    __syncthreads();

    const int lane = threadIdx.x & 31;
    const int wv   = threadIdx.x >> 5;
    const int tile = blockIdx.x * 4 + wv;
    const int ntiles = (N + 15) / 16;
    if (tile >= ntiles) return;
    const int tb   = tile * 16;
    const int rl   = lane & 15;
    const int half = lane >> 4;
    const int kb   = half * 2;
    const int arow = min(tb + rl, N - 1);
    const bool full = (tb + 16 <= N);

    for (int i = lane; i < 16 * 144; i += 32) {
        const int row = i / 144, col = i % 144;
        const int node = min(tb + row, N - 1);
        lnode[wv][i] = (col < 128) ? nf[node * 128 + col] : attrs[node * 16 + (col - 128)];
    }
    __builtin_amdgcn_wave_barrier();

    {   // scalar channel
        v8f cm0 = {}, cm1 = {}, cs0 = {}, cs1 = {};
        #pragma unroll 4
        for (int j = 0; j < 16; ++j) {            // ns @ W2s'
            const int k0 = 4 * j + kb;
            const int kp = (k0 >> 1) * 64;
            v2f a  = ld2(&acc[(size_t)arow * 256 + k0]);
            v2f b0 = ld2(&lw2s[kp + rl * 2]);
            v2f b1 = ld2(&lw2s[kp + (16 + rl) * 2]);
            cm0 = wmma4(a, b0, cm0); cm1 = wmma4(a, b1, cm1);
        }
        #pragma unroll 4
        for (int j = 0; j < 128; ++j) {           // (s_in (x) attrs) @ Wsc_s', K=512
            const int k0 = 4 * j + kb;
            const int kp = (k0 >> 1) * 64;
            const int u0 = k0 >> 4, a0 = k0 & 15;
            const int u1 = (k0 + 1) >> 4, a1 = (k0 + 1) & 15;
            v2f a;
            a.x = lnode[wv][rl * 144 + u0] * lnode[wv][rl * 144 + 128 + a0];
            a.y = lnode[wv][rl * 144 + u1] * lnode[wv][rl * 144 + 128 + a1];
            v2f b0 = ld2(&wsp_s[kp + rl * 2]);
            v2f b1 = ld2(&wsp_s[kp + (16 + rl) * 2]);
            cs0 = wmma4(a, b0, cs0); cs1 = wmma4(a, b1, cs1);
        }
        if (full) {
            #pragma unroll
            for (int r = 0; r < 8; ++r) {
                const int node = tb + r + 8 * half;
                out[node * 128 + rl]      = cm0[r] + cs0[r];
                out[node * 128 + 16 + rl] = cm1[r] + cs1[r];
            }
        } else {
            #pragma unroll
            for (int r = 0; r < 8; ++r) {
                const int node = tb + r + 8 * half;
                if (node < N) {
                    out[node * 128 + rl]      = cm0[r] + cs0[r];
                    out[node * 128 + 16 + rl] = cm1[r] + cs1[r];
                }
            }
        }
    }
    #pragma unroll
    for (int cc = 0; cc < 3; ++cc) {              // vector channels
        v8f cm0 = {}, cm1 = {}, cs0 = {}, cs1 = {};
        #pragma unroll 4
        for (int j = 0; j < 16; ++j) {            // nv[c] @ W2v'
            const int k0 = 4 * j + kb;
            const int kp = (k0 >> 1) * 64;
            v2f a;
            a.x = acc[(size_t)arow * 256 + 64 + k0 * 3 + cc];
            a.y = acc[(size_t)arow * 256 + 64 + (k0 + 1) * 3 + cc];
            v2f b0 = ld2(&lw2v[kp + rl * 2]);
            v2f b1 = ld2(&lw2v[kp + (16 + rl) * 2]);
            cm0 = wmma4(a, b0, cm0); cm1 = wmma4(a, b1, cm1);
        }
        #pragma unroll 4
        for (int j = 0; j < 128; ++j) {           // (v_in[c] (x) attrs) @ Wsc_v', K=512
            const int k0 = 4 * j + kb;
            const int kp = (k0 >> 1) * 64;
            const int u0 = k0 >> 4, a0 = k0 & 15;
            const int u1 = (k0 + 1) >> 4, a1 = (k0 + 1) & 15;
            v2f a;
            a.x = lnode[wv][rl * 144 + 32 + u0 * 3 + cc] * lnode[wv][rl * 144 + 128 + a0];
            a.y = lnode[wv][rl * 144 + 32 + u1 * 3 + cc] * lnode[wv][rl * 144 + 128 + a1];
            v2f b0 = ld2(&wsp_v[kp + rl * 2]);
            v2f b1 = ld2(&wsp_v[kp + (16 + rl) * 2]);
            cs0 = wmma4(a, b0, cs0); cs1 = wmma4(a, b1, cs1);
        }
        if (full) {
            #pragma unroll
            for (int r = 0; r < 8; ++r) {
                const int node = tb + r + 8 * half;
                out[node * 128 + 32 + rl * 3 + cc]        = cm0[r] + cs0[r];
                out[node * 128 + 32 + (16 + rl) * 3 + cc] = cm1[r] + cs1[r];
            }
        } else {
            #pragma unroll
            for (int r = 0; r < 8; ++r) {
                const int node = tb + r + 8 * half;
                if (node < N) {
                    out[node * 128 + 32 + rl * 3 + cc]        = cm0[r] + cs0[r];
                    out[node * 128 + 32 + (16 + rl) * 3 + cc] = cm1[r] + cs1[r];
                }
            }
        }
    }
}

// ---------------------------------------------------------------------------
extern "C" void kernel_launch(void* const* d_in, const int* in_sizes, int n_in,
                              void* d_out, int out_size, void* d_ws, size_t ws_size,
                              hipStream_t stream) {
    const float* nf    = (const float*)d_in[0];
    const float* attrs = (const float*)d_in[1];
    const float* emb   = (const float*)d_in[2];
    const float* eattr = (const float*)d_in[3];
    const int*   eidx  = (const int*)d_in[4];
    const float* W1s   = (const float*)d_in[5];
    const float* W1v   = (const float*)d_in[6];
    const float* Wr0   = (const float*)d_in[7];
    const float* Wr1   = (const float*)d_in[8];
    const float* W2s   = (const float*)d_in[9];
    const float* W2v   = (const float*)d_in[10];
    const float* Wsc_s = (const float*)d_in[11];
    const float* Wsc_v = (const float*)d_in[12];
    float* out = (float*)d_out;

    const int N = in_sizes[0] / 128;   // node_feats is N x 4M
    const int E = in_sizes[2] / 8;     // edge_embedding is E x R

    float* sv    = (float*)d_ws;                        // N x 128
    float* acc   = sv + (size_t)N * 128;                // N x 256 (ns | nv)
    float* wsp_s = acc + (size_t)N * 256;               // 512x32 pair layout
    float* wsp_v = wsp_s + 512 * 32;

    hipMemsetAsync(acc, 0, (size_t)N * 256 * sizeof(float), stream);

    k_prep_wsc<<<64, 256, 0, stream>>>(Wsc_s, Wsc_v, wsp_s, wsp_v);

    const int ntilesN = (N + 15) / 16;
    const int blkN = (ntilesN + 3) / 4;
    k_node_pre<<<blkN, 128, 0, stream>>>(nf, W1s, W1v, sv, N);

    k_edge<<<1024, 128, 0, stream>>>(emb, eattr, eidx, Wr0, Wr1, sv, acc, E);

    k_node_post<<<blkN, 128, 0, stream>>>(nf, attrs, acc, W2s, W2v, wsp_s, wsp_v, out, N);
}